// GetRelation_75857712382391
// MI455X (gfx1250) — compile-verified
//
#include <hip/hip_runtime.h>
#include <math.h>

typedef __attribute__((ext_vector_type(2))) float v2f;
typedef __attribute__((ext_vector_type(8))) float v8f;

#define NN 1024
#define TT 8
#define DD 64
#define KK 8

__device__ __forceinline__ v8f wmma_f32_4(v2f a, v2f b, v8f c) {
  // D = A(16x4 f32) * B(4x16 f32) + C(16x16 f32)
  return __builtin_amdgcn_wmma_f32_16x16x4_f32(false, a, false, b, (short)0, c,
                                               false, false);
}

// ---------------------------------------------------------------------------
// prep: HA[i] = x[:,i,:] @ W1m[:64]  + b1m      (8 x 1024 x 64)
//       HB[i] = x[:,i,:] @ W1m[64:]             (8 x 1024 x 64)
//       HSA   = x[:,7,:] @ W1s[:64]  + b1s      (1024 x 64)
//       HSB   = x[:,7,:] @ W1s[64:]             (1024 x 64)
//       asoft = softmax(a)                       (8)
// One wave per block; block computes a 16-row x 64-col output tile via WMMA.
// grid = (64 row tiles, 18 jobs)
// ---------------------------------------------------------------------------
__global__ __launch_bounds__(32)
void prep_kernel(const float* __restrict__ x, const float* __restrict__ W1m,
                 const float* __restrict__ b1m, const float* __restrict__ W1s,
                 const float* __restrict__ b1s, const float* __restrict__ a,
                 float* __restrict__ HA, float* __restrict__ HB,
                 float* __restrict__ HSA, float* __restrict__ HSB,
                 float* __restrict__ asoft)
{
  const int j  = blockIdx.y;
  const int P0 = blockIdx.x * 16;
  const int l  = threadIdx.x;
  const int m  = l & 15;   // row (A) / col (B,C) within tile
  const int kh = l >> 4;   // K-half selector

  // softmax(a) done once by a single lane
  if (blockIdx.x == 0 && j == 0 && l == 0) {
    float mx = a[0];
    #pragma unroll
    for (int i = 1; i < TT; ++i) mx = fmaxf(mx, a[i]);
    float s = 0.f;
    float e[TT];
    #pragma unroll
    for (int i = 0; i < TT; ++i) { e[i] = __expf(a[i] - mx); s += e[i]; }
    #pragma unroll
    for (int i = 0; i < TT; ++i) asoft[i] = e[i] / s;
  }

  const float* W;  const float* bias;  float* outp;  int it;
  if (j < 8)        { it = j;     W = W1m;           bias = b1m;    outp = HA + (size_t)j * (NN * DD); }
  else if (j < 16)  { it = j - 8; W = W1m + 64 * DD; bias = nullptr; outp = HB + (size_t)(j - 8) * (NN * DD); }
  else if (j == 16) { it = 7;     W = W1s;           bias = b1s;    outp = HSA; }
  else              { it = 7;     W = W1s + 64 * DD; bias = nullptr; outp = HSB; }

  v8f acc[4];
  #pragma unroll
  for (int ct = 0; ct < 4; ++ct) acc[ct] = (v8f){0.f,0.f,0.f,0.f,0.f,0.f,0.f,0.f};

  #pragma unroll
  for (int kk = 0; kk < 16; ++kk) {
    const int kd = kk * 4 + kh * 2;
    // A frag: A[m][kd] = x[P0+m, it, kd] ; float2 of consecutive K
    const float* xp = x + (size_t)(P0 + m) * (TT * DD) + it * DD + kd;
    v2f av; av.x = xp[0]; av.y = xp[1];
    #pragma unroll
    for (int ct = 0; ct < 4; ++ct) {
      // B frag: B[kd][n] = W[kd][ct*16+n] ; K-adjacent rows stride DD apart
      const float* wp = W + (size_t)kd * DD + ct * 16 + m;
      v2f bv; bv.x = wp[0]; bv.y = wp[DD];
      acc[ct] = wmma_f32_4(av, bv, acc[ct]);
    }
  }

  // C/D layout: lane<16 -> M=r, lane>=16 -> M=8+r ; N = (lane&15)+ct*16
  #pragma unroll
  for (int ct = 0; ct < 4; ++ct) {
    const int col = ct * 16 + m;
    const float bv = bias ? bias[col] : 0.f;
    #pragma unroll
    for (int r = 0; r < 8; ++r) {
      const int row = kh * 8 + r;
      outp[(size_t)(P0 + row) * DD + col] = acc[ct][r] + bv;
    }
  }
}

// ---------------------------------------------------------------------------
// pairs: one 16x16 (p,q) tile per 256-thread block (8 waves).
// Phase A: wave w computes D_w[p][q] = sum_c E[w,p,c]*E[w,q,c] via 16 WMMA
//          f32 16x16x4 steps (two independent accumulator chains) -> LDS
//          in [pair][k] layout for b128 readback.
// Phase B: one (p,q) pair per thread, all LDS traffic as ds_load_b128.
// ---------------------------------------------------------------------------
__global__ __launch_bounds__(256)
void pairs_kernel(const float* __restrict__ Ind, const float* __restrict__ Loc,
                  const float* __restrict__ E,   const float* __restrict__ g,
                  const float* __restrict__ W2m, const float* __restrict__ b2m,
                  const float* __restrict__ W2s, const float* __restrict__ b2s,
                  const float* __restrict__ HA,  const float* __restrict__ HB,
                  const float* __restrict__ HSA, const float* __restrict__ HSB,
                  const float* __restrict__ asoft, float* __restrict__ out)
{
  const int P0 = blockIdx.x * 16;
  const int Q0 = blockIdx.y * 16;
  const int tid  = threadIdx.x;
  const int wave = tid >> 5;
  const int lane = tid & 31;

  __shared__ float Dsh[16 * 16 * KK];          // [pair p*16+q][k]
  __shared__ float sSA[16 * 68];               // HSA tile (padded stride 68)
  __shared__ float sSB[16 * 68];               // HSB tile
  __shared__ float sA[16 * 68];                // per-i HA tile
  __shared__ float sB[16 * 68];                // per-i HB tile
  __shared__ float sW2s[DD * KK];              // 64 x 8
  __shared__ float sw2m[DD];
  __shared__ float sasoft[TT];
  __shared__ float sb2s[KK];

  // ---- cooperative persistent loads (float4 per thread) ----
  {
    const int row = tid >> 4;
    const int c4  = (tid & 15) * 4;
    *(float4*)&sSA[row * 68 + c4] = *(const float4*)&HSA[(size_t)(P0 + row) * DD + c4];
    *(float4*)&sSB[row * 68 + c4] = *(const float4*)&HSB[(size_t)(Q0 + row) * DD + c4];
  }
  if (tid < 128) *(float4*)&sW2s[tid * 4] = *(const float4*)&W2s[tid * 4];
  if (tid < 16)  *(float4*)&sw2m[tid * 4] = *(const float4*)&W2m[tid * 4];
  if (tid < TT)  sasoft[tid] = asoft[tid];
  if (tid < KK)  sb2s[tid]   = b2s[tid];

  // ---- Phase A: D_k tile via WMMA f32 16x16x4, wave 'wave' handles k=wave
  {
    const int m  = lane & 15;
    const int kh = lane >> 4;
    v8f acc0 = (v8f){0.f,0.f,0.f,0.f,0.f,0.f,0.f,0.f};
    v8f acc1 = (v8f){0.f,0.f,0.f,0.f,0.f,0.f,0.f,0.f};
    const float* Ek = E + (size_t)wave * NN * DD;
    #pragma unroll
    for (int kk = 0; kk < 16; kk += 2) {
      const int kd0 = kk * 4 + kh * 2;
      const int kd1 = (kk + 1) * 4 + kh * 2;
      const float* ea0 = Ek + (size_t)(P0 + m) * DD + kd0;
      const float* eb0 = Ek + (size_t)(Q0 + m) * DD + kd0;
      const float* ea1 = Ek + (size_t)(P0 + m) * DD + kd1;
      const float* eb1 = Ek + (size_t)(Q0 + m) * DD + kd1;
      v2f av0; av0.x = ea0[0]; av0.y = ea0[1];
      v2f bv0; bv0.x = eb0[0]; bv0.y = eb0[1];
      v2f av1; av1.x = ea1[0]; av1.y = ea1[1];
      v2f bv1; bv1.x = eb1[0]; bv1.y = eb1[1];
      acc0 = wmma_f32_4(av0, bv0, acc0);
      acc1 = wmma_f32_4(av1, bv1, acc1);
    }
    #pragma unroll
    for (int r = 0; r < 8; ++r) {
      const int row = kh * 8 + r;
      Dsh[(row * 16 + m) * KK + wave] = acc0[r] + acc1[r];
    }
  }
  __syncthreads();

  // ---- Phase B: one pair per thread ----
  const int p  = tid >> 4;
  const int q  = tid & 15;
  const int gp = P0 + p;
  const int gq = Q0 + q;
  const float b2m0 = b2m[0];

  // A_mkt: 8 time steps, tiles staged through LDS, b128 LDS reads
  float amkt = 0.f;
  for (int i = 0; i < TT; ++i) {
    {
      const int row = tid >> 4;
      const int c4  = (tid & 15) * 4;
      *(float4*)&sA[row * 68 + c4] =
          *(const float4*)&HA[(size_t)i * NN * DD + (size_t)(P0 + row) * DD + c4];
      *(float4*)&sB[row * 68 + c4] =
          *(const float4*)&HB[(size_t)i * NN * DD + (size_t)(Q0 + row) * DD + c4];
    }
    __syncthreads();
    float s = 0.f;
    #pragma unroll
    for (int c = 0; c < DD; c += 4) {
      const float4 va = *(const float4*)&sA[p * 68 + c];
      const float4 vb = *(const float4*)&sB[q * 68 + c];
      const float4 vw = *(const float4*)&sw2m[c];
      s = fmaf(fmaxf(va.x + vb.x, 0.f), vw.x, s);
      s = fmaf(fmaxf(va.y + vb.y, 0.f), vw.y, s);
      s = fmaf(fmaxf(va.z + vb.z, 0.f), vw.z, s);
      s = fmaf(fmaxf(va.w + vb.w, 0.f), vw.w, s);
    }
    amkt = fmaf(fmaxf(s + b2m0, 0.f), sasoft[i], amkt);
    __syncthreads();
  }

  // logits: hs relu-gated 64 -> 8, b128 LDS reads
  float lg[KK];
  #pragma unroll
  for (int k = 0; k < KK; ++k) lg[k] = sb2s[k];
  #pragma unroll
  for (int c = 0; c < DD; c += 4) {
    const float4 va = *(const float4*)&sSA[p * 68 + c];
    const float4 vb = *(const float4*)&sSB[q * 68 + c];
    const float hv[4] = { fmaxf(va.x + vb.x, 0.f), fmaxf(va.y + vb.y, 0.f),
                          fmaxf(va.z + vb.z, 0.f), fmaxf(va.w + vb.w, 0.f) };
    #pragma unroll
    for (int cc = 0; cc < 4; ++cc) {
      const float4 w0 = *(const float4*)&sW2s[(c + cc) * KK];
      const float4 w1 = *(const float4*)&sW2s[(c + cc) * KK + 4];
      lg[0] = fmaf(hv[cc], w0.x, lg[0]);
      lg[1] = fmaf(hv[cc], w0.y, lg[1]);
      lg[2] = fmaf(hv[cc], w0.z, lg[2]);
      lg[3] = fmaf(hv[cc], w0.w, lg[3]);
      lg[4] = fmaf(hv[cc], w1.x, lg[4]);
      lg[5] = fmaf(hv[cc], w1.y, lg[5]);
      lg[6] = fmaf(hv[cc], w1.z, lg[6]);
      lg[7] = fmaf(hv[cc], w1.w, lg[7]);
    }
  }

  // gumbel: two b128 global loads
  const float4 g0 = *(const float4*)&g[((size_t)gp * NN + gq) * KK];
  const float4 g1 = *(const float4*)&g[((size_t)gp * NN + gq) * KK + 4];
  const float gk[KK] = { g0.x, g0.y, g0.z, g0.w, g1.x, g1.y, g1.z, g1.w };
  #pragma unroll
  for (int k = 0; k < KK; ++k) lg[k] = fmaxf(lg[k], 0.f) + gk[k];

  // D values for this pair: two b128 LDS reads
  const float4 d0 = *(const float4*)&Dsh[(p * 16 + q) * KK];
  const float4 d1 = *(const float4*)&Dsh[(p * 16 + q) * KK + 4];
  const float dk[KK] = { d0.x, d0.y, d0.z, d0.w, d1.x, d1.y, d1.z, d1.w };

  // softmax over K=8 fused with D-weighted sum
  float mx = lg[0];
  #pragma unroll
  for (int k = 1; k < KK; ++k) mx = fmaxf(mx, lg[k]);
  float se = 0.f, dacc = 0.f;
  #pragma unroll
  for (int k = 0; k < KK; ++k) {
    const float e = __expf(lg[k] - mx);
    se += e;
    dacc = fmaf(e, dk[k], dacc);
  }
  const float anobs = dacc / se;

  // output stack [Ind, Loc, A_mkt, A_nobs] : (4, 1024, 1024)
  const size_t idx = (size_t)gp * NN + gq;
  const size_t N2  = (size_t)NN * NN;
  out[idx]          = Ind[idx];
  out[N2 + idx]     = Loc[idx];
  out[2 * N2 + idx] = amkt;
  out[3 * N2 + idx] = anobs;
}

// ---------------------------------------------------------------------------
extern "C" void kernel_launch(void* const* d_in, const int* in_sizes, int n_in,
                              void* d_out, int out_size, void* d_ws, size_t ws_size,
                              hipStream_t stream) {
  (void)in_sizes; (void)n_in; (void)out_size; (void)ws_size;
  const float* x   = (const float*)d_in[0];
  const float* Ind = (const float*)d_in[1];
  const float* Loc = (const float*)d_in[2];
  const float* a   = (const float*)d_in[3];
  const float* W1m = (const float*)d_in[4];
  const float* b1m = (const float*)d_in[5];
  const float* W2m = (const float*)d_in[6];
  const float* b2m = (const float*)d_in[7];
  const float* W1s = (const float*)d_in[8];
  const float* b1s = (const float*)d_in[9];
  const float* W2s = (const float*)d_in[10];
  const float* b2s = (const float*)d_in[11];
  const float* E   = (const float*)d_in[12];
  const float* g   = (const float*)d_in[13];
  float* out = (float*)d_out;

  float* ws   = (float*)d_ws;
  float* HA   = ws;                                   // 8*1024*64
  float* HB   = HA + (size_t)TT * NN * DD;            // 8*1024*64
  float* HSA  = HB + (size_t)TT * NN * DD;            // 1024*64
  float* HSB  = HSA + (size_t)NN * DD;                // 1024*64
  float* asoft= HSB + (size_t)NN * DD;                // 8

  prep_kernel<<<dim3(NN / 16, 18), dim3(32), 0, stream>>>(
      x, W1m, b1m, W1s, b1s, a, HA, HB, HSA, HSB, asoft);

  pairs_kernel<<<dim3(NN / 16, NN / 16), dim3(256), 0, stream>>>(
      Ind, Loc, E, g, W2m, b2m, W2s, b2s, HA, HB, HSA, HSB, asoft, out);
}